// PureSSM_28913719837239
// MI455X (gfx1250) — compile-verified
//
#include <hip/hip_runtime.h>
#include <math.h>

typedef __attribute__((ext_vector_type(2))) float v2f;
typedef __attribute__((ext_vector_type(4))) float v4f;
typedef __attribute__((ext_vector_type(8))) float v8f;

#define DMODEL  768
#define DSTATE  16
#define DCONV   4
#define DINNER  1536
#define DTRANK  48
#define BATCH   4
#define SEQLEN  2048
#define NTOK    (BATCH * SEQLEN)          /* 8192 */
#define XDBLW   (DTRANK + 2 * DSTATE)     /* 80   */
#define NCHUNK  32
#define CLEN    (SEQLEN / NCHUNK)         /* 64   */

// ---------------------------------------------------------------------------
// fp32 WMMA GEMM:  C[M,N] = act( A[M,K] * W[N,K]^T (+ bias) )
// Wave tile: MT x NT 16x16 tiles, K step 4 via v_wmma_f32_16x16x4_f32,
// software-pipelined fragment loads (double buffer).
// Fragment layouts per CDNA5 ISA 7.12.2.  ACT: 0 = none, 1 = softplus(+bias)
// ---------------------------------------------------------------------------
template <int MT, int NT, int ACT>
__global__ __launch_bounds__(256) void gemm_wmma_f32(
    const float* __restrict__ A, int lda,
    const float* __restrict__ W, int ldb,
    const float* __restrict__ bias,
    float* __restrict__ C, int ldc,
    int MtGroups, int NtGroups, int K)
{
    const int lane = threadIdx.x & 31;
    const int w    = blockIdx.x * (blockDim.x >> 5) + (threadIdx.x >> 5);
    const int mg   = w / NtGroups;
    const int ng   = w % NtGroups;
    if (mg >= MtGroups) return;               // wave-uniform exit

    const int m0   = mg * MT * 16;
    const int n0   = ng * NT * 16;
    const int r15  = lane & 15;
    const int koff = (lane >> 4) << 1;        // 0 or 2

    const float* arow[MT];
    const float* brow[NT];
#pragma unroll
    for (int mi = 0; mi < MT; ++mi)
        arow[mi] = A + (size_t)(m0 + mi * 16 + r15) * lda + koff;
#pragma unroll
    for (int j = 0; j < NT; ++j)
        brow[j] = W + (size_t)(n0 + j * 16 + r15) * ldb + koff;

    v8f acc[MT][NT];
#pragma unroll
    for (int mi = 0; mi < MT; ++mi)
#pragma unroll
        for (int j = 0; j < NT; ++j)
            acc[mi][j] = (v8f){0.f, 0.f, 0.f, 0.f, 0.f, 0.f, 0.f, 0.f};

    v2f a_cur[MT], b_cur[NT];
#pragma unroll
    for (int mi = 0; mi < MT; ++mi) a_cur[mi] = *(const v2f*)(arow[mi]);
#pragma unroll
    for (int j = 0; j < NT; ++j)   b_cur[j]  = *(const v2f*)(brow[j]);

    for (int k = 0; k < K; k += 4) {
        const int kn = (k + 4 < K) ? (k + 4) : k;   // clamp on last iter
        v2f a_nxt[MT], b_nxt[NT];
#pragma unroll
        for (int mi = 0; mi < MT; ++mi) a_nxt[mi] = *(const v2f*)(arow[mi] + kn);
#pragma unroll
        for (int j = 0; j < NT; ++j)   b_nxt[j]  = *(const v2f*)(brow[j] + kn);

#pragma unroll
        for (int mi = 0; mi < MT; ++mi)
#pragma unroll
            for (int j = 0; j < NT; ++j)
                acc[mi][j] = __builtin_amdgcn_wmma_f32_16x16x4_f32(
                    false, a_cur[mi], false, b_cur[j], (short)0,
                    acc[mi][j], false, false);

#pragma unroll
        for (int mi = 0; mi < MT; ++mi) a_cur[mi] = a_nxt[mi];
#pragma unroll
        for (int j = 0; j < NT; ++j)   b_cur[j]  = b_nxt[j];
    }

    const int rOff = (lane >> 4) * 8;
#pragma unroll
    for (int mi = 0; mi < MT; ++mi) {
#pragma unroll
        for (int j = 0; j < NT; ++j) {
            const int col = n0 + j * 16 + r15;
#pragma unroll
            for (int r = 0; r < 8; ++r) {
                float v = acc[mi][j][r];
                if (ACT == 1) {
                    v += bias[col];
                    v = (v > 20.f) ? v : log1pf(__expf(v));   // softplus
                }
                C[(size_t)(m0 + mi * 16 + r + rOff) * ldc + col] = v;
            }
        }
    }
}

// ---------------------------------------------------------------------------
// Depthwise causal conv (k=4) + bias + SiLU:  xb = xz[..., :DINNER]
// ---------------------------------------------------------------------------
__global__ __launch_bounds__(256) void conv_silu_kernel(
    const float* __restrict__ xz, const float* __restrict__ conv_w,
    const float* __restrict__ conv_b, float* __restrict__ xc)
{
    int idx = blockIdx.x * blockDim.x + threadIdx.x;
    if (idx >= NTOK * DINNER) return;
    const int d = idx % DINNER;
    const int t = idx / DINNER;
    const int l = t % SEQLEN;

    float acc = conv_b[d];
#pragma unroll
    for (int j = 0; j < DCONV; ++j) {
        const int ll = l - (DCONV - 1) + j;
        if (ll >= 0)
            acc += xz[(size_t)(t - (DCONV - 1) + j) * (2 * DINNER) + d] *
                   conv_w[d * DCONV + j];
    }
    xc[(size_t)t * DINNER + d] = acc / (1.f + __expf(-acc));   // silu
}

// ---------------------------------------------------------------------------
// Chunk-parallel selective scan.  Recurrence h_t = exp(dt_t*A)h_{t-1} + dt_t
// x_t B_t is linear & diagonal, so chunks compose:  h_end = P*h_begin + hloc,
// with P_n = prod_t exp(dt_t A_n) = exp(A_n * sum_t dt_t).
// Thread order (b, chunk, d): consecutive d -> coalesced x/dt, uniform B/C.
// ---------------------------------------------------------------------------
__global__ __launch_bounds__(256) void scan_pass1(
    const float* __restrict__ xc, const float* __restrict__ dt,
    const float* __restrict__ xdbl, const float* __restrict__ A_log,
    float* __restrict__ hloc, float* __restrict__ sbuf)
{
    const int tid = blockIdx.x * blockDim.x + threadIdx.x;
    if (tid >= BATCH * NCHUNK * DINNER) return;
    const int d = tid % DINNER;
    const int c = (tid / DINNER) % NCHUNK;
    const int b = tid / (DINNER * NCHUNK);

    float Arow[DSTATE], h[DSTATE];
#pragma unroll
    for (int n = 0; n < DSTATE; ++n) {
        Arow[n] = -__expf(A_log[d * DSTATE + n]);
        h[n] = 0.f;
    }
    float S = 0.f;

    for (int l = c * CLEN; l < (c + 1) * CLEN; ++l) {
        const size_t t   = (size_t)b * SEQLEN + l;
        const float x_t  = xc[t * DINNER + d];
        const float dt_t = dt[t * DINNER + d];
        const float* Bp  = xdbl + t * XDBLW + DTRANK;
        v4f Bv[4];
#pragma unroll
        for (int q = 0; q < 4; ++q) Bv[q] = *(const v4f*)(Bp + 4 * q);
        const float dx = dt_t * x_t;
#pragma unroll
        for (int n = 0; n < DSTATE; ++n)
            h[n] = __expf(dt_t * Arow[n]) * h[n] + dx * Bv[n >> 2][n & 3];
        S += dt_t;
    }

    const size_t base = (size_t)tid;          // (b*NCHUNK + c)*DINNER + d
    sbuf[base] = S;
#pragma unroll
    for (int n = 0; n < DSTATE; ++n) hloc[base * DSTATE + n] = h[n];
}

__global__ __launch_bounds__(256) void scan_pass2(
    const float* __restrict__ hloc, const float* __restrict__ sbuf,
    const float* __restrict__ A_log, float* __restrict__ hinit)
{
    const int tid = blockIdx.x * blockDim.x + threadIdx.x;
    if (tid >= BATCH * DINNER) return;
    const int d = tid % DINNER;
    const int b = tid / DINNER;

    float Arow[DSTATE], h[DSTATE];
#pragma unroll
    for (int n = 0; n < DSTATE; ++n) {
        Arow[n] = -__expf(A_log[d * DSTATE + n]);
        h[n] = 0.f;
    }
    for (int c = 0; c < NCHUNK; ++c) {
        const size_t base = ((size_t)b * NCHUNK + c) * DINNER + d;
#pragma unroll
        for (int n = 0; n < DSTATE; ++n) hinit[base * DSTATE + n] = h[n];
        const float S = sbuf[base];
#pragma unroll
        for (int n = 0; n < DSTATE; ++n)
            h[n] = __expf(Arow[n] * S) * h[n] + hloc[base * DSTATE + n];
    }
}

__global__ __launch_bounds__(256) void scan_pass3(
    const float* __restrict__ xc, const float* __restrict__ dt,
    const float* __restrict__ xdbl, const float* __restrict__ xz,
    const float* __restrict__ A_log, const float* __restrict__ D_param,
    const float* __restrict__ hinit, float* __restrict__ ybuf)
{
    const int tid = blockIdx.x * blockDim.x + threadIdx.x;
    if (tid >= BATCH * NCHUNK * DINNER) return;
    const int d = tid % DINNER;
    const int c = (tid / DINNER) % NCHUNK;
    const int b = tid / (DINNER * NCHUNK);

    float Arow[DSTATE], h[DSTATE];
    const size_t base = (size_t)tid;
#pragma unroll
    for (int n = 0; n < DSTATE; ++n) {
        Arow[n] = -__expf(A_log[d * DSTATE + n]);
        h[n] = hinit[base * DSTATE + n];
    }
    const float Dp = D_param[d];

    for (int l = c * CLEN; l < (c + 1) * CLEN; ++l) {
        const size_t t   = (size_t)b * SEQLEN + l;
        const float x_t  = xc[t * DINNER + d];
        const float dt_t = dt[t * DINNER + d];
        const float* Bp  = xdbl + t * XDBLW + DTRANK;
        const float* Cp  = Bp + DSTATE;
        v4f Bv[4], Cv[4];
#pragma unroll
        for (int q = 0; q < 4; ++q) {
            Bv[q] = *(const v4f*)(Bp + 4 * q);
            Cv[q] = *(const v4f*)(Cp + 4 * q);
        }
        const float dx = dt_t * x_t;
        float y = 0.f;
#pragma unroll
        for (int n = 0; n < DSTATE; ++n) {
            h[n] = __expf(dt_t * Arow[n]) * h[n] + dx * Bv[n >> 2][n & 3];
            y += h[n] * Cv[n >> 2][n & 3];
        }
        y += Dp * x_t;
        const float z = xz[t * (2 * DINNER) + DINNER + d];
        ybuf[t * DINNER + d] = y * (z / (1.f + __expf(-z)));
    }
}

// ---------------------------------------------------------------------------
extern "C" void kernel_launch(void* const* d_in, const int* in_sizes, int n_in,
                              void* d_out, int out_size, void* d_ws, size_t ws_size,
                              hipStream_t stream)
{
    const float* x       = (const float*)d_in[0];
    const float* W_in    = (const float*)d_in[1];
    const float* conv_w  = (const float*)d_in[2];
    const float* conv_b  = (const float*)d_in[3];
    const float* W_x     = (const float*)d_in[4];
    const float* W_dt    = (const float*)d_in[5];
    const float* b_dt    = (const float*)d_in[6];
    const float* A_log   = (const float*)d_in[7];
    const float* D_param = (const float*)d_in[8];
    const float* W_out   = (const float*)d_in[9];
    float* out = (float*)d_out;

    float* ws    = (float*)d_ws;
    float* xz    = ws;                        size_t o = (size_t)NTOK * 2 * DINNER;
    float* xc    = ws + o;                    o += (size_t)NTOK * DINNER;
    float* xdbl  = ws + o;                    o += (size_t)NTOK * XDBLW;
    float* dt    = ws + o;                    o += (size_t)NTOK * DINNER;
    float* ybuf  = ws + o;                    o += (size_t)NTOK * DINNER;
    float* hloc  = ws + o;                    o += (size_t)BATCH * NCHUNK * DINNER * DSTATE;
    float* hinit = ws + o;                    o += (size_t)BATCH * NCHUNK * DINNER * DSTATE;
    float* sbuf  = ws + o;                    o += (size_t)BATCH * NCHUNK * DINNER;
    (void)ws_size; (void)in_sizes; (void)n_in; (void)out_size;

    const int WPB = 8;  // waves per 256-thread block

    // 1) xz = x @ W_in^T   (8192 x 3072 x 768), wave tile 32x64
    {
        const int Mg = NTOK / 32, Ng = (2 * DINNER) / 64;
        gemm_wmma_f32<2, 4, 0><<<Mg * Ng / WPB, 256, 0, stream>>>(
            x, DMODEL, W_in, DMODEL, nullptr, xz, 2 * DINNER, Mg, Ng, DMODEL);
    }
    // 2) depthwise conv + silu -> xc
    conv_silu_kernel<<<(NTOK * DINNER) / 256, 256, 0, stream>>>(xz, conv_w, conv_b, xc);
    // 3) x_dbl = xc @ W_x^T   (8192 x 80 x 1536), whole N (5 tiles) per wave
    {
        const int Mg = NTOK / 32, Ng = 1;
        gemm_wmma_f32<2, 5, 0><<<Mg * Ng / WPB, 256, 0, stream>>>(
            xc, DINNER, W_x, DINNER, nullptr, xdbl, XDBLW, Mg, Ng, DINNER);
    }
    // 4) dt = softplus(dt_r @ W_dt^T + b_dt)   (8192 x 1536 x 48), lda = 80
    {
        const int Mg = NTOK / 32, Ng = DINNER / 64;
        gemm_wmma_f32<2, 4, 1><<<Mg * Ng / WPB, 256, 0, stream>>>(
            xdbl, XDBLW, W_dt, DTRANK, b_dt, dt, DINNER, Mg, Ng, DTRANK);
    }
    // 5) chunk-parallel selective scan + gate -> ybuf
    scan_pass1<<<(BATCH * NCHUNK * DINNER) / 256, 256, 0, stream>>>(
        xc, dt, xdbl, A_log, hloc, sbuf);
    scan_pass2<<<(BATCH * DINNER) / 256, 256, 0, stream>>>(
        hloc, sbuf, A_log, hinit);
    scan_pass3<<<(BATCH * NCHUNK * DINNER) / 256, 256, 0, stream>>>(
        xc, dt, xdbl, xz, A_log, D_param, hinit, ybuf);
    // 6) out = y @ W_out^T   (8192 x 768 x 1536)
    {
        const int Mg = NTOK / 32, Ng = DMODEL / 64;
        gemm_wmma_f32<2, 4, 0><<<Mg * Ng / WPB, 256, 0, stream>>>(
            ybuf, DINNER, W_out, DINNER, nullptr, out, DMODEL, Mg, Ng, DINNER);
    }
}